// PSROIPool_64493228917124
// MI455X (gfx1250) — compile-verified
//
#include <hip/hip_runtime.h>

// Position-sensitive ROI pooling (R-FCN), k=7, H=W=1024.
// Inputs: d_in[0] = x (49*1024*1024 f32), d_in[1] = region (4 f32: ci, cj, h, w)
// Output: d_out[0] = scalar f32 region score.
//
// One wave32 block per bin (49 blocks). Each block derives its bin's integer
// pixel rectangle from `region` (same linspace/floor/ceil math as the
// reference), prefetches the rectangle's rows (global_prefetch, no counter),
// accumulates f32 partials with a 4-deep unrolled row loop (4 independent
// loads in flight per iteration), then reduces the 64 per-lane partials with
// V_WMMA_F32_16X16X4_F32 (A = partials 16x4, B = ones 4x16) + one shfl_xor.

#define KK 7
#define HH 1024
#define WW 1024

typedef float v2f __attribute__((ext_vector_type(2)));
typedef float v8f __attribute__((ext_vector_type(8)));

__global__ void __launch_bounds__(32)
psroi_bin_kernel(const float* __restrict__ x,
                 const float* __restrict__ region,
                 float* __restrict__ bin_means) {
    const int bin  = blockIdx.x;       // 0..48, channel c = bi*7 + bj
    const int bi   = bin / KK;
    const int bj   = bin % KK;
    const int lane = threadIdx.x;      // 0..31 (wave32)

    // ---- bin extent (mirrors the reference's linspace + floor/ceil) ----
    const float ci = region[0], cj = region[1];
    const float h  = region[2], w  = region[3];
    const float i0 = ci - 0.5f * h;
    const float j0 = cj - 0.5f * w;
    // interior point (bi+1) of a (k+2)-point linspace over [i0, i0+h]
    const float ic = i0 + (float)(bi + 1) * (h / (float)(KK + 1));
    const float jc = j0 + (float)(bj + 1) * (w / (float)(KK + 1));
    const float bh = h / (float)KK;
    const float bw = w / (float)KK;

    int r0 = (int)floorf((ic - 0.5f * bh) * (float)HH);
    int r1 = (int)ceilf ((ic + 0.5f * bh) * (float)HH);
    int c0 = (int)floorf((jc - 0.5f * bw) * (float)WW);
    int c1 = (int)ceilf ((jc + 0.5f * bw) * (float)WW);
    r0 = max(r0, 0); r1 = min(r1, HH);
    c0 = max(c0, 0); c1 = min(c1, WW);
    const int nrows = r1 - r0;
    const int ncols = c1 - c0;
    const long total = (long)max(nrows, 0) * (long)max(ncols, 0);

    const float* __restrict__ base = x + (size_t)bin * (size_t)(HH * WW);

    // ---- non-blocking prefetch of the rectangle (global_prefetch_b8):
    // one lane per row; fetches whole cachelines, warms WGP$/L2, no counter.
    for (int r = r0 + lane; r < r1; r += 32) {
        __builtin_prefetch(&base[(size_t)r * WW + c0], 0, 3);
    }

    // ---- per-lane partial sums. Lanes stride columns; rows unrolled x4 so
    // four independent global_load_b32 are in flight per wait.
    float a0 = 0.0f, a1 = 0.0f, a2 = 0.0f, a3 = 0.0f;
    for (int c = c0 + lane; c < c1; c += 32) {
        const float* __restrict__ colp = base + c;
        int r = r0;
        for (; r + 3 < r1; r += 4) {
            const float v0 = colp[(size_t)(r + 0) * WW];
            const float v1 = colp[(size_t)(r + 1) * WW];
            const float v2 = colp[(size_t)(r + 2) * WW];
            const float v3 = colp[(size_t)(r + 3) * WW];
            a0 += v0; a1 += v1; a2 += v2; a3 += v3;
        }
        for (; r < r1; ++r) {
            a0 += colp[(size_t)r * WW];
        }
    }
    const float acc0 = a0 + a2;
    const float acc1 = a1 + a3;

    // ---- wave reduction via the matrix unit:
    // A(16x4) = per-lane partials, B(4x16) = ones  =>  D[m,n] = sum_k A[m,k]
    v2f a; a[0] = acc0; a[1] = acc1;
    v2f b; b[0] = 1.0f; b[1] = 1.0f;
    v8f cmat = {0.f, 0.f, 0.f, 0.f, 0.f, 0.f, 0.f, 0.f};
    v8f d = __builtin_amdgcn_wmma_f32_16x16x4_f32(
        /*neg_a=*/false, a, /*neg_b=*/false, b,
        /*c_mod=*/(short)0, cmat, /*reuse_a=*/false, /*reuse_b=*/false);

    // lanes 0-15 hold D rows M=0..7 at N=lane; lanes 16-31 hold M=8..15.
    float s = d[0] + d[1] + d[2] + d[3] + d[4] + d[5] + d[6] + d[7];
    s += __shfl_xor(s, 16, 32);        // full 16x4 sum in every lane

    if (lane == 0) {
        const float cnt = (total > 0) ? (float)total : 1.0f;
        bin_means[bin] = s / cnt;
    }
}

__global__ void __launch_bounds__(32)
psroi_finalize_kernel(const float* __restrict__ bin_means,
                      float* __restrict__ out) {
    if (threadIdx.x == 0 && blockIdx.x == 0) {
        float s = 0.0f;
        for (int i = 0; i < KK * KK; ++i) s += bin_means[i];
        out[0] = s * (1.0f / (float)(KK * KK));
    }
}

extern "C" void kernel_launch(void* const* d_in, const int* in_sizes, int n_in,
                              void* d_out, int out_size, void* d_ws, size_t ws_size,
                              hipStream_t stream) {
    (void)in_sizes; (void)n_in; (void)out_size; (void)ws_size;
    const float* x      = (const float*)d_in[0];
    const float* region = (const float*)d_in[1];
    float* out          = (float*)d_out;
    float* bin_means    = (float*)d_ws;   // 49 f32 scratch

    psroi_bin_kernel<<<KK * KK, 32, 0, stream>>>(x, region, bin_means);
    psroi_finalize_kernel<<<1, 32, 0, stream>>>(bin_means, out);
}